// GuideFilter2d_46566035423318
// MI455X (gfx1250) — compile-verified
//
#include <hip/hip_runtime.h>

#define HH   512
#define WW   512
#define TILE 32
#define INW  36   // input tile width  (TILE + 2*2 halo)
#define ABW  34   // a/b tile width    (TILE + 2*1 halo)
#define EPS  0.01f
#define NTHREADS 256

// ---- CDNA5 async global->LDS copy (GLOBAL_LOAD_ASYNC_TO_LDS_B32, ASYNCcnt) ----
__device__ __forceinline__ void async_load_f32(const float* gptr, unsigned lds_byte_off) {
    // vdst = VGPR holding LDS byte offset, vaddr = 64-bit global address, saddr = off
    asm volatile("global_load_async_to_lds_b32 %0, %1, off"
                 :: "v"(lds_byte_off), "v"(gptr)
                 : "memory");
}

__device__ __forceinline__ void wait_async0() {
#if __has_builtin(__builtin_amdgcn_s_wait_asynccnt)
    __builtin_amdgcn_s_wait_asynccnt(0);
#else
    asm volatile("s_wait_asynccnt 0" ::: "memory");
#endif
}

// low 32 bits of a flat LDS pointer are the wave-relative LDS byte offset (ISA 10.2)
__device__ __forceinline__ unsigned lds_off(const void* p) {
    return (unsigned)(unsigned long long)p;
}

// 1 / (number of valid taps of a radius-1 window at coord c in [0,n))
__device__ __forceinline__ float inv3(int c, int n) {
    return (c == 0 || c == n - 1) ? 0.5f : (1.0f / 3.0f);
}

__global__ __launch_bounds__(NTHREADS)
void guided_filter_fused(const float* __restrict__ gI_,
                         const float* __restrict__ gP_,
                         float* __restrict__ gO_) {
    __shared__ float sI [INW * INW];
    __shared__ float sP [INW * INW];
    __shared__ float hI [INW * ABW];
    __shared__ float hP [INW * ABW];
    __shared__ float hII[INW * ABW];
    __shared__ float hIP[INW * ABW];
    __shared__ float sA [ABW * ABW];
    __shared__ float sB [ABW * ABW];
    __shared__ float hA [ABW * TILE];
    __shared__ float hB [ABW * TILE];

    const int t = threadIdx.x;
    const long long pbase = (long long)blockIdx.z * (HH * WW);
    const float* gI = gI_ + pbase;
    const float* gP = gP_ + pbase;
    float*       gO = gO_ + pbase;

    const int gx0 = blockIdx.x * TILE - 2;
    const int gy0 = blockIdx.y * TILE - 2;

    // ---------------- Stage A: stage I,p (36x36 with halo) into LDS ----------------
    const bool interior = (gx0 >= 0) && (gy0 >= 0) &&
                          (gx0 + INW <= WW) && (gy0 + INW <= HH);
    if (interior) {
        // async DMA-style copies, no VGPR staging; tracked by ASYNCcnt
        for (int idx = t; idx < INW * INW; idx += NTHREADS) {
            int iy = idx / INW, ix = idx - iy * INW;
            long long g = (long long)(gy0 + iy) * WW + (gx0 + ix);
            async_load_f32(gI + g, lds_off(&sI[idx]));
            async_load_f32(gP + g, lds_off(&sP[idx]));
        }
        wait_async0();
    } else {
        for (int idx = t; idx < INW * INW; idx += NTHREADS) {
            int iy = idx / INW, ix = idx - iy * INW;
            int gy = gy0 + iy, gx = gx0 + ix;
            bool v = (unsigned)gy < (unsigned)HH && (unsigned)gx < (unsigned)WW;
            float vi = 0.f, vp = 0.f;
            if (v) {
                long long g = (long long)gy * WW + gx;
                vi = gI[g];
                vp = gP[g];
            }
            sI[idx] = vi;
            sP[idx] = vp;
        }
    }
    __syncthreads();

    // ------- Stage B1: horizontal 3-sums of I, p, I*I, I*p (36 rows x 34 cols) -------
    for (int idx = t; idx < INW * ABW; idx += NTHREADS) {
        int iy = idx / ABW, jx = idx - iy * ABW;   // center input col = jx+1
        int b = iy * INW + jx;
        float i0 = sI[b], i1 = sI[b + 1], i2 = sI[b + 2];
        float p0 = sP[b], p1 = sP[b + 1], p2 = sP[b + 2];
        hI [idx] = i0 + i1 + i2;
        hP [idx] = p0 + p1 + p2;
        hII[idx] = i0 * i0 + i1 * i1 + i2 * i2;
        hIP[idx] = i0 * p0 + i1 * p1 + i2 * p2;
    }
    __syncthreads();

    // ------- Stage B2: vertical 3-sums -> means -> a,b  (34x34, halo 1) -------
    for (int idx = t; idx < ABW * ABW; idx += NTHREADS) {
        int iy = idx / ABW, jx = idx - iy * ABW;   // center input coords (iy+1, jx+1)
        int gy = gy0 + iy + 1, gx = gx0 + jx + 1;
        float a = 0.f, b = 0.f;                    // out-of-image a,b contribute 0
        if ((unsigned)gy < (unsigned)HH && (unsigned)gx < (unsigned)WW) {
            int r0 = iy * ABW + jx;
            float sIv = hI [r0] + hI [r0 + ABW] + hI [r0 + 2 * ABW];
            float sPv = hP [r0] + hP [r0 + ABW] + hP [r0 + 2 * ABW];
            float sII = hII[r0] + hII[r0 + ABW] + hII[r0 + 2 * ABW];
            float sIP = hIP[r0] + hIP[r0 + ABW] + hIP[r0 + 2 * ABW];
            float invn = inv3(gy, HH) * inv3(gx, WW);
            float mI = sIv * invn, mP = sPv * invn;
            float varI = sII * invn - mI * mI;
            float cov  = sIP * invn - mI * mP;
            a = cov / (varI + EPS);
            b = mP - a * mI;
        }
        sA[idx] = a;
        sB[idx] = b;
    }
    __syncthreads();

    // ------- Stage C1: horizontal 3-sums of a,b (34 rows x 32 cols) -------
    for (int idx = t; idx < ABW * TILE; idx += NTHREADS) {
        int iy = idx / TILE, ox = idx - iy * TILE; // center a-col = ox+1
        int b = iy * ABW + ox;
        hA[idx] = sA[b] + sA[b + 1] + sA[b + 2];
        hB[idx] = sB[b] + sB[b + 1] + sB[b + 2];
    }
    __syncthreads();

    // ------- Stage C2: vertical 3-sums -> mean_a, mean_b -> output (32x32) -------
    for (int idx = t; idx < TILE * TILE; idx += NTHREADS) {
        int oy = idx >> 5, ox = idx & 31;
        int r0 = oy * TILE + ox;
        float sAv = hA[r0] + hA[r0 + TILE] + hA[r0 + 2 * TILE];
        float sBv = hB[r0] + hB[r0 + TILE] + hB[r0 + 2 * TILE];
        int gy = gy0 + oy + 2, gx = gx0 + ox + 2;  // always in-image
        float invn = inv3(gy, HH) * inv3(gx, WW);
        float ma = sAv * invn, mb = sBv * invn;
        float Iv = sI[(oy + 2) * INW + (ox + 2)];
        gO[(long long)gy * WW + gx] = ma * Iv + mb;
    }
}

extern "C" void kernel_launch(void* const* d_in, const int* in_sizes, int n_in,
                              void* d_out, int out_size, void* d_ws, size_t ws_size,
                              hipStream_t stream) {
    const float* I = (const float*)d_in[0];
    const float* P = (const float*)d_in[1];
    float*       O = (float*)d_out;
    (void)d_ws; (void)ws_size; (void)n_in; (void)out_size;

    int planes = in_sizes[0] / (HH * WW);   // B*C = 128 for the reference shape
    dim3 grid((WW + TILE - 1) / TILE, (HH + TILE - 1) / TILE, planes);
    guided_filter_fused<<<grid, NTHREADS, 0, stream>>>(I, P, O);
}